// reblurWithKernel_20882130993766
// MI455X (gfx1250) — compile-verified
//
#include <hip/hip_runtime.h>
#include <stdint.h>

#define PAD    16
#define KW     33
#define IMG    512
#define PIMG   544              // 512 + 2*16
#define NHB    128
#define NWB    128
#define PLANE  (PIMG * PIMG)    // 295936
#define KPLANE (NHB * NWB)      // 16384

#define HBWG   2                // block rows per workgroup
#define WQWG   4                // block-quads per workgroup (each quad = 4 blocks)
#define TPB    (32 * HBWG * WQWG)     // 256 threads = 8 wave32, one quad per wave
#define TROWS  (4 * HBWG + 32)  // 40 padded-image rows per tile
#define TCOLS  (16 * WQWG + 32) // 96 padded-image cols per tile
#define TILE_F (3 * TROWS * TCOLS)    // 11520 floats
#define STG_F  (2 * 36 * 4)           // per-wave kernel stage: 2 bufs x 36 v x 4 wb = 288 floats
#define LDS_F  (TILE_F + 8 * STG_F)   // 13824 floats = 55296 B

typedef unsigned int u32;
typedef unsigned long long u64;
typedef __attribute__((ext_vector_type(2))) float v2f;
typedef __attribute__((ext_vector_type(8))) float v8f;
typedef __attribute__((ext_vector_type(4))) u32 tdm_v4u;
typedef __attribute__((ext_vector_type(8))) int tdm_v8i;
typedef __attribute__((ext_vector_type(4))) int tdm_v4i;

#if defined(__has_builtin)
#if __has_builtin(__builtin_amdgcn_tensor_load_to_lds) && __has_builtin(__builtin_amdgcn_s_wait_tensorcnt)
#define USE_TDM 1
#endif
#endif

// ---------------- Pass 1: edge-replicate pad into workspace -----------------
__global__ void reblur_pad_kernel(const float* __restrict__ img,
                                  float* __restrict__ pimg) {
    int i = blockIdx.x * blockDim.x + threadIdx.x;
    const int total = 3 * PLANE;
    if (i >= total) return;
    int c  = i / PLANE;
    int r  = i - c * PLANE;
    int hp = r / PIMG;
    int wp = r - hp * PIMG;
    int h = hp - PAD; h = h < 0 ? 0 : (h > IMG - 1 ? IMG - 1 : h);
    int w = wp - PAD; w = w < 0 ? 0 : (w > IMG - 1 ? IMG - 1 : w);
    pimg[i] = img[((u64)c * IMG + h) * IMG + w];
}

// ---------------- Pass 2: WMMA banded-GEMM per block-quad -------------------
// One wave computes D[16x16] = sum over (u,s) of
//   A[(c,oi),(u,s)] = padded[c, 4hb+oi+u, 16*wquad+s]        (raw image tile)
//   B[(u,s),(wbl,oj)] = K[u, s-(4wbl+oj), hb, 4wquad+wbl]    (banded kernels)
// giving all 48 outputs of 4 adjacent blocks in D rows 0..11.
__global__ __launch_bounds__(TPB) void reblur_wmma_kernel(
    const float* __restrict__ pimg,   // padded image [3][544][544]
    const float* __restrict__ kern,   // [1089][128][128]
    float* __restrict__ out)          // [3][512][512]
{
    extern __shared__ float lds[];    // [TILE_F] image tile + 8 x [STG_F] kernel stages

    const int tid   = threadIdx.x;
    const int wv    = tid >> 5;               // 0..7
    const int lane  = tid & 31;
    const int hi    = lane >> 4;              // lane half (k-split of fragments)
    const int n     = lane & 15;              // GEMM column: (wbl, oj)
    const int wql   = wv & 3;                 // quad within WG
    const int prw   = wv >> 2;                // hb row within WG
    const int hb0   = blockIdx.y * HBWG;
    const int hb    = hb0 + prw;
    const int wquad = blockIdx.x * WQWG + wql;

    // ---- Stage A: image tile (3 x TROWS x TCOLS fp32) into LDS via TDM ----
#ifdef USE_TDM
    if (wv == 0) {
        const u64 gaddr = (u64)(uintptr_t)(pimg + (u64)(4 * hb0) * PIMG
                                                + (u64)(16 * WQWG) * blockIdx.x);
        const u32 laddr = (u32)(uintptr_t)(&lds[0]);

        tdm_v4u g0;
        g0[0] = 1u;                                      // count=1, user descriptor
        g0[1] = laddr;                                   // lds_addr (bytes)
        g0[2] = (u32)(gaddr & 0xffffffffu);
        g0[3] = (u32)((gaddr >> 32) & 0x01ffffffu) | (2u << 30);  // addr[56:32], type=2

        tdm_v8i g1;
        g1[0] = (int)(2u << 16);                                    // data_size = 4B
        g1[1] = (int)(((u32)PIMG & 0xffffu) << 16);                 // tensor_dim0 lo16
        g1[2] = (int)((((u32)PIMG >> 16) & 0xffffu)
                    | (((u32)PIMG & 0xffffu) << 16));               // dim0 hi | dim1 lo
        g1[3] = (int)((((u32)PIMG >> 16) & 0xffffu)
                    | ((u32)TCOLS << 16));                          // dim1 hi | tile_dim0
        g1[4] = (int)((u32)TROWS | (3u << 16));                     // tile_dim1 | tile_dim2
        g1[5] = (int)(u32)PIMG;                                     // dim0_stride lo32
        g1[6] = (int)(((u32)PLANE & 0xffffu) << 16);                // dim0_str hi | dim1_str lo
        g1[7] = (int)((u32)PLANE >> 16);                            // dim1_stride[47:16]

        tdm_v4i g2; g2[0] = 3; g2[1] = 0; g2[2] = 0; g2[3] = 0;     // tensor_dim2 = 3
        tdm_v4i g3; g3[0] = 0; g3[1] = 0; g3[2] = 0; g3[3] = 0;

#if __clang_major__ >= 23
        tdm_v8i g4 = {0, 0, 0, 0, 0, 0, 0, 0};
        __builtin_amdgcn_tensor_load_to_lds(g0, g1, g2, g3, g4, 0);
#else
        __builtin_amdgcn_tensor_load_to_lds(g0, g1, g2, g3, 0);
#endif
        __builtin_amdgcn_s_wait_tensorcnt(0);
    }
    __syncthreads();
#else
    for (int i = tid; i < TILE_F; i += TPB) {
        const int c  = i / (TROWS * TCOLS);
        const int r2 = i - c * (TROWS * TCOLS);
        const int rr = r2 / TCOLS;
        const int cc = r2 - rr * TCOLS;
        lds[i] = pimg[((u64)c * PIMG + (4 * hb0 + rr)) * PIMG
                      + ((u64)(16 * WQWG) * blockIdx.x + cc)];
    }
    __syncthreads();
#endif

    // ---- per-lane loop-invariant setup ----
    const int stageF = TILE_F + wv * STG_F;   // this wave's private kernel stage
    const int wbl    = n >> 2;                // block within quad
    int mc = n >> 2; mc = mc > 2 ? 2 : mc;    // A row channel (rows 12..15 duplicate c=2)
    const int aIdx = (mc * TROWS + 4 * prw + (n & 3)) * TCOLS + 16 * wql + 2 * hi;

    // zero the "invalid shift" slots (v=33..35) of both stage buffers, once
    if (lane < 24) {
        const int buf = lane / 12, sl = lane - buf * 12;
        lds[stageF + buf * 144 + 132 + sl] = 0.0f;
    }

    // B-element LDS indices for buf0; invalid shifts -> zeroed slot (v=35)
    int bidx0[12], bidx1[12];
#pragma unroll
    for (int q = 0; q < 12; ++q) {
        const int v1 = 4 * q + hi - n;        // kernel col for B row k = hi
        const int v2 = v1 + 2;                //                 ... k = hi+2
        bidx0[q] = stageF + (((u32)v1 < 33u ? v1 : 35) * 4 + wbl);
        bidx1[q] = stageF + (((u32)v2 < 33u ? v2 : 35) * 4 + wbl);
    }

    // Kernel-row staging, split so the loadcnt wait lands AFTER a compute block:
    //   fetch_row:   global -> VGPR (issued before compute, no wait)
    //   publish_row: VGPR -> LDS    (placed after compute; wait hidden by WMMAs)
    const int kcolBase = hb * NWB + wquad * 4;
    auto fetch_row = [&](int u, float4& kv, float4& kv2) {
        kv  = *(const float4*)(kern + (u64)(u * KW + lane) * KPLANE + kcolBase);
        // v=32 quad is lane-invariant (uniform address): load unconditionally
        kv2 = *(const float4*)(kern + (u64)(u * KW + 32) * KPLANE + kcolBase);
    };
    auto publish_row = [&](int buf, const float4& kv, const float4& kv2) {
        *(float4*)&lds[stageF + (buf * 36 + lane) * 4] = kv;
        if (lane == 0)
            *(float4*)&lds[stageF + (buf * 36 + 32) * 4] = kv2;
    };

    v8f acc = {0.f, 0.f, 0.f, 0.f, 0.f, 0.f, 0.f, 0.f};

    // 12 chained f32 WMMAs consume one kernel row u (K-chunks s0 = 4q)
    auto compute_u = [&](int u, const int bufF) {
        const float* arow = &lds[aIdx + u * TCOLS];
#pragma unroll
        for (int q = 0; q < 12; ++q) {
            v2f a = *(const v2f*)(arow + 4 * q);
            v2f b;
            b[0] = lds[bidx0[q] + bufF];
            b[1] = lds[bidx1[q] + bufF];
            acc = __builtin_amdgcn_wmma_f32_16x16x4_f32(
                false, a, false, b, (short)0, acc, false, false);
        }
    };

    // ---- Stage B: software-pipelined u-loop over double-buffered stages ----
    float4 kva, kvb;
    fetch_row(0, kva, kvb);
    publish_row(0, kva, kvb);
#pragma unroll 1
    for (int u = 0; u < KW; u += 2) {
        if (u + 1 < KW) fetch_row(u + 1, kva, kvb);   // issue loads for u+1
        compute_u(u, 0);                              // ... hidden behind row u
        if (u + 1 < KW) {
            publish_row(1, kva, kvb);                 // wait lands here, post-compute
            if (u + 2 < KW) fetch_row(u + 2, kva, kvb);
            compute_u(u + 1, 144);
            if (u + 2 < KW) publish_row(0, kva, kvb);
        }
    }

    // ---- Stage C: scatter D rows 0..11 -> out[c, 4hb+oi, 16*wquad+n] ----
#pragma unroll
    for (int r = 0; r < 8; ++r) {
        const int m = r + 8 * hi;             // D row held by this lane/VGPR
        if (m < 12) {
            out[((u64)((m >> 2) * IMG + 4 * hb + (m & 3))) * IMG + 16 * wquad + n] = acc[r];
        }
    }
}

extern "C" void kernel_launch(void* const* d_in, const int* in_sizes, int n_in,
                              void* d_out, int out_size, void* d_ws, size_t ws_size,
                              hipStream_t stream) {
    (void)in_sizes; (void)n_in; (void)out_size; (void)ws_size;
    const float* img  = (const float*)d_in[0];   // (1,3,512,512) f32
    const float* kern = (const float*)d_in[1];   // (1,1089,128,128) f32
    float* outp = (float*)d_out;                 // (1,3,512,512) f32
    float* pimg = (float*)d_ws;                  // 3*544*544 f32 = 3.55 MB scratch

    const int ptotal = 3 * PLANE;
    reblur_pad_kernel<<<(ptotal + 255) / 256, 256, 0, stream>>>(img, pimg);

    dim3 grid((NWB / 4) / WQWG, NHB / HBWG);     // (8, 64) = 512 WGs, 4096 waves
    const size_t lds_bytes = (size_t)LDS_F * sizeof(float);   // 55296 B
    reblur_wmma_kernel<<<grid, TPB, lds_bytes, stream>>>(pimg, kern, outp);
}